// ModuleCLSPool_23278722744350
// MI455X (gfx1250) — compile-verified
//
#include <hip/hip_runtime.h>
#include <math.h>

typedef __attribute__((ext_vector_type(2))) float v2f;
typedef __attribute__((ext_vector_type(8))) float v8f;

static constexpr int B_ = 16, V_ = 4096, D_ = 512, K_ = 64, H_ = 8;
static constexpr int DK_ = 64, DE_ = 16, DFF_ = 2048;

#define USE_ASYNC_LDS 1

__device__ __forceinline__ v8f wmma_f32(v2f a, v2f b, v8f c) {
  // V_WMMA_F32_16X16X4_F32 : D = A(16x4) * B(4x16) + C(16x16)
  return __builtin_amdgcn_wmma_f32_16x16x4_f32(false, a, false, b, (short)0, c,
                                               false, false);
}

#if USE_ASYNC_LDS
// GV-mode async copy of 16 bytes global -> LDS (per active lane).
// LDS address = low 32 bits of the generic pointer (aperture offset).
__device__ __forceinline__ void async_b128(void* lds_ptr, const void* gptr) {
  unsigned lds_off = (unsigned)(unsigned long long)lds_ptr;
  asm volatile("global_load_async_to_lds_b128 %0, %1, off"
               :
               : "v"(lds_off), "v"(gptr)
               : "memory");
}
__device__ __forceinline__ void wait_async0() {
  asm volatile("s_wait_asynccnt 0x0" ::: "memory");
}
#endif

// ---------------------------------------------------------------------------
// cls = cls0 + E @ E2D_w + E2D_b   (batch independent, [K,D])
// qn  = LayerNorm(cls)
// ---------------------------------------------------------------------------
__global__ __launch_bounds__(256) void cls_ln_kernel(
    const float* __restrict__ E, const float* __restrict__ E2Dw,
    const float* __restrict__ E2Db, const float* __restrict__ cls0,
    const float* __restrict__ lng, const float* __restrict__ lnb,
    float* __restrict__ cls, float* __restrict__ qn) {
  __shared__ float Erow[DE_];
  __shared__ float red[256];
  const int k = blockIdx.x, tid = threadIdx.x;
  if (tid < DE_) Erow[tid] = E[k * DE_ + tid];
  __syncthreads();
  float vals[2];
#pragma unroll
  for (int j = 0; j < 2; ++j) {
    const int d = tid + j * 256;
    float acc = cls0[k * D_ + d] + E2Db[d];
#pragma unroll
    for (int e = 0; e < DE_; ++e) acc += Erow[e] * E2Dw[e * D_ + d];
    vals[j] = acc;
    cls[k * D_ + d] = acc;
  }
  red[tid] = vals[0] + vals[1];
  __syncthreads();
  for (int st = 128; st > 0; st >>= 1) {
    if (tid < st) red[tid] += red[tid + st];
    __syncthreads();
  }
  const float mu = red[0] / (float)D_;
  __syncthreads();
  red[tid] = (vals[0] - mu) * (vals[0] - mu) + (vals[1] - mu) * (vals[1] - mu);
  __syncthreads();
  for (int st = 128; st > 0; st >>= 1) {
    if (tid < st) red[tid] += red[tid + st];
    __syncthreads();
  }
  const float rstd = rsqrtf(red[0] / (float)D_ + 1e-5f);
#pragma unroll
  for (int j = 0; j < 2; ++j) {
    const int d = tid + j * 256;
    qn[k * D_ + d] = (vals[j] - mu) * rstd * lng[d] + lnb[d];
  }
}

// ---------------------------------------------------------------------------
// Row-wise LayerNorm for [rows, 512]; one block per row.
// ---------------------------------------------------------------------------
__global__ __launch_bounds__(256) void ln_rows_kernel(
    const float* __restrict__ X, const float* __restrict__ g,
    const float* __restrict__ bta, float* __restrict__ Y) {
  __shared__ float red[256];
  const int row = blockIdx.x, tid = threadIdx.x;
  const float v0 = X[(size_t)row * D_ + tid];
  const float v1 = X[(size_t)row * D_ + tid + 256];
  red[tid] = v0 + v1;
  __syncthreads();
  for (int st = 128; st > 0; st >>= 1) {
    if (tid < st) red[tid] += red[tid + st];
    __syncthreads();
  }
  const float mu = red[0] / (float)D_;
  __syncthreads();
  red[tid] = (v0 - mu) * (v0 - mu) + (v1 - mu) * (v1 - mu);
  __syncthreads();
  for (int st = 128; st > 0; st >>= 1) {
    if (tid < st) red[tid] += red[tid + st];
    __syncthreads();
  }
  const float rstd = rsqrtf(red[0] / (float)D_ + 1e-5f);
  Y[(size_t)row * D_ + tid] = (v0 - mu) * rstd * g[tid] + bta[tid];
  Y[(size_t)row * D_ + tid + 256] =
      (v1 - mu) * rstd * g[tid + 256] + bta[tid + 256];
}

// ---------------------------------------------------------------------------
// count_per_k[b,k] = sum_v assign[b,v,k]; empty = count <= 1e-6
// ---------------------------------------------------------------------------
__global__ __launch_bounds__(256) void count_kernel(
    const float* __restrict__ assign, float* __restrict__ cnt,
    float* __restrict__ emp_out, float* __restrict__ emp_ws) {
  __shared__ float red[4][64];
  const int b = blockIdx.x;
  const int slot = threadIdx.x & 63, grp = threadIdx.x >> 6;
  float s = 0.f;
  for (int v = grp; v < V_; v += 4)
    s += assign[((size_t)b * V_ + v) * K_ + slot];
  red[grp][slot] = s;
  __syncthreads();
  if (threadIdx.x < 64) {
    const float t = red[0][slot] + red[1][slot] + red[2][slot] + red[3][slot];
    cnt[b * K_ + slot] = t;
    const float e = (t <= 1e-6f) ? 1.f : 0.f;
    emp_out[b * K_ + slot] = e;
    emp_ws[b * K_ + slot] = e;
  }
}

// ---------------------------------------------------------------------------
// fp32 WMMA GEMM: C[M,N] = epi(A[M,K] @ W[K,N] + bias [, R])
//   EPI 0: none   1: exact gelu   2: + residual R[M,N]
// Block tile 128x64, BK=16, 256 threads / 8 waves.
// Wave w: rows (w>>1)*32..+31 (two 16-row groups), cols (w&1)*32..+31.
// Weight tile stored transposed (Bt[n][k]) so every fragment is a
// contiguous, 8B-aligned LDS pair -> single ds_load_b64, no repacking.
// Requires N%64==0, K%16==0; M handled with row guards.
// ---------------------------------------------------------------------------
template <int EPI>
__global__ __launch_bounds__(256) void gemm_f32_wmma(
    const float* __restrict__ A, const float* __restrict__ W,
    const float* __restrict__ bias, const float* __restrict__ R,
    float* __restrict__ C, int M, int N, int Kd) {
  __shared__ float As[128][20];  // pad 20 -> 16B-aligned rows, conflict-free
  __shared__ float Bt[64][18];   // pad 18 -> 8B-aligned pairs, conflict-free
  const int tid = threadIdx.x;
  const int lane = tid & 31, wave = tid >> 5;
  const int half = lane >> 4, l16 = lane & 15;
  const int bm = blockIdx.y * 128, bn = blockIdx.x * 64;
  const int wr = wave >> 1, wc = wave & 1;

  v8f c00 = {}, c01 = {}, c10 = {}, c11 = {};

  for (int k0 = 0; k0 < Kd; k0 += 16) {
    // A tile: 128x16 floats as 512 x 16B chunks (async global->LDS)
    for (int j = tid; j < 512; j += 256) {
      const int r = j >> 2, qc = (j & 3) * 4;
      if (bm + r < M) {
#if USE_ASYNC_LDS
        async_b128(&As[r][qc], A + (size_t)(bm + r) * Kd + k0 + qc);
#else
        const float* g = A + (size_t)(bm + r) * Kd + k0 + qc;
        As[r][qc] = g[0]; As[r][qc + 1] = g[1];
        As[r][qc + 2] = g[2]; As[r][qc + 3] = g[3];
#endif
      }
    }
    // W tile transposed: Bt[n][k]
    for (int i = tid; i < 16 * 64; i += 256) {
      const int r = i >> 6, c = i & 63;
      Bt[c][r] = W[(size_t)(k0 + r) * N + (bn + c)];
    }
#if USE_ASYNC_LDS
    wait_async0();
#endif
    __syncthreads();
#pragma unroll
    for (int kk = 0; kk < 16; kk += 4) {
      const int ko = kk + 2 * half;  // lanes 0-15: K={kk,kk+1}; 16-31: {kk+2,kk+3}
      const v2f a0 = *(const v2f*)&As[wr * 32 + l16][ko];
      const v2f a1 = *(const v2f*)&As[wr * 32 + 16 + l16][ko];
      const v2f b0 = *(const v2f*)&Bt[wc * 32 + l16][ko];
      const v2f b1 = *(const v2f*)&Bt[wc * 32 + 16 + l16][ko];
      c00 = wmma_f32(a0, b0, c00);
      c01 = wmma_f32(a0, b1, c01);
      c10 = wmma_f32(a1, b0, c10);
      c11 = wmma_f32(a1, b1, c11);
    }
    __syncthreads();
  }
#pragma unroll
  for (int i = 0; i < 8; ++i) {
    // C/D layout: VGPR i -> M = i (+8 for lanes 16-31), N = lane&15
    const int row0 = bm + wr * 32 + i + half * 8;
    const int row1 = row0 + 16;
    const int col0 = bn + wc * 32 + l16;
    const int col1 = col0 + 16;
    const float bb0 = bias[col0], bb1 = bias[col1];
    float o00 = c00[i] + bb0, o01 = c01[i] + bb1;
    float o10 = c10[i] + bb0, o11 = c11[i] + bb1;
    if (EPI == 1) {
      o00 = 0.5f * o00 * (1.0f + erff(o00 * 0.7071067811865476f));
      o01 = 0.5f * o01 * (1.0f + erff(o01 * 0.7071067811865476f));
      o10 = 0.5f * o10 * (1.0f + erff(o10 * 0.7071067811865476f));
      o11 = 0.5f * o11 * (1.0f + erff(o11 * 0.7071067811865476f));
    }
    if (row0 < M) {
      if (EPI == 2) {
        o00 += R[(size_t)row0 * N + col0];
        o01 += R[(size_t)row0 * N + col1];
      }
      C[(size_t)row0 * N + col0] = o00;
      C[(size_t)row0 * N + col1] = o01;
    }
    if (row1 < M) {
      if (EPI == 2) {
        o10 += R[(size_t)row1 * N + col0];
        o11 += R[(size_t)row1 * N + col1];
      }
      C[(size_t)row1 * N + col0] = o10;
      C[(size_t)row1 * N + col1] = o11;
    }
  }
}

// ---------------------------------------------------------------------------
// Flash attention, one workgroup per (b,h). 64-wide V chunks.
// S = q@k^T/8 + log(clip(assign)+1e-6); online softmax; acc += P@V.
// ks[] is reused as the probability buffer after S is formed.
// vst[] holds the V chunk transposed so P@V fragments are contiguous pairs.
// ---------------------------------------------------------------------------
__global__ __launch_bounds__(256) void attn_flash_kernel(
    const float* __restrict__ q,       // [K_, D_] (batch independent)
    const float* __restrict__ kbuf,    // [B_, V_, D_]
    const float* __restrict__ vbuf,    // [B_, V_, D_]
    const float* __restrict__ assign,  // [B_, V_, K_]
    float* __restrict__ mout) {        // [B_, K_, D_]
  __shared__ float qs[64][68];   // pad 68 -> 16B-aligned rows
  __shared__ float ks[64][68];   // k chunk, then probs
  __shared__ float vst[64][68];  // v chunk transposed: vst[d][v]
  __shared__ float rmax[64], rsum[64], rcorr[64];

  const int b = blockIdx.x >> 3, h = blockIdx.x & 7;
  const int tid = threadIdx.x;
  const int lane = tid & 31, wave = tid >> 5;
  const int half = lane >> 4, l16 = lane & 15;
  const int tr = wave >> 1, tc0 = (wave & 1) * 2;
  const float scale = 0.125f;  // 1/sqrt(DK)

  // q tile (row-major, contiguous both sides -> async)
  for (int j = tid; j < 64 * 16; j += 256) {
    const int r = j >> 4, qc = (j & 15) * 4;
#if USE_ASYNC_LDS
    async_b128(&qs[r][qc], q + (size_t)r * D_ + h * DK_ + qc);
#else
    const float* g = q + (size_t)r * D_ + h * DK_ + qc;
    qs[r][qc] = g[0]; qs[r][qc + 1] = g[1];
    qs[r][qc + 2] = g[2]; qs[r][qc + 3] = g[3];
#endif
  }
  if (tid < 64) {
    rmax[tid] = -3.4e38f;
    rsum[tid] = 0.f;
  }

  v8f a0 = {}, a1 = {};  // accumulator tiles over (slot, d)

  for (int v0 = 0; v0 < V_; v0 += 64) {
    __syncthreads();  // previous chunk fully consumed; init visible
    for (int j = tid; j < 64 * 16; j += 256) {
      const int r = j >> 4, qc = (j & 15) * 4;
#if USE_ASYNC_LDS
      async_b128(&ks[r][qc],
                 kbuf + ((size_t)b * V_ + (v0 + r)) * D_ + h * DK_ + qc);
#else
      const float* g = kbuf + ((size_t)b * V_ + (v0 + r)) * D_ + h * DK_ + qc;
      ks[r][qc] = g[0]; ks[r][qc + 1] = g[1];
      ks[r][qc + 2] = g[2]; ks[r][qc + 3] = g[3];
#endif
    }
    // v chunk transposed (scatter -> regular loads)
    for (int i = tid; i < 64 * 64; i += 256) {
      const int r = i >> 6, c = i & 63;
      vst[c][r] = vbuf[((size_t)b * V_ + (v0 + r)) * D_ + h * DK_ + c];
    }
#if USE_ASYNC_LDS
    wait_async0();
#endif
    __syncthreads();
    // S = q @ k^T   (contract over d; ks rows give contiguous B pairs)
    v8f s0 = {}, s1 = {};
#pragma unroll
    for (int kk = 0; kk < 64; kk += 4) {
      const int ko = kk + 2 * half;
      const v2f a = *(const v2f*)&qs[tr * 16 + l16][ko];
      const v2f b0 = *(const v2f*)&ks[tc0 * 16 + l16][ko];
      const v2f b1 = *(const v2f*)&ks[tc0 * 16 + 16 + l16][ko];
      s0 = wmma_f32(a, b0, s0);
      s1 = wmma_f32(a, b1, s1);
    }
    __syncthreads();  // everyone finished reading ks
    // masked scores -> ks (probs buffer)
#pragma unroll
    for (int i = 0; i < 8; ++i) {
      const int row = tr * 16 + i + half * 8;  // slot
      const int c0i = tc0 * 16 + l16;          // v within chunk
      const int c1i = c0i + 16;
      float m0 = assign[((size_t)b * V_ + (v0 + c0i)) * K_ + row];
      float m1 = assign[((size_t)b * V_ + (v0 + c1i)) * K_ + row];
      m0 = logf(fminf(fmaxf(m0, 0.f), 1.f) + 1e-6f);
      m1 = logf(fminf(fmaxf(m1, 0.f), 1.f) + 1e-6f);
      ks[row][c0i] = s0[i] * scale + m0;
      ks[row][c1i] = s1[i] * scale + m1;
    }
    __syncthreads();
    // online softmax (one thread per slot row)
    if (tid < 64) {
      const float mold = rmax[tid];
      float cm = mold;
      for (int c = 0; c < 64; ++c) cm = fmaxf(cm, ks[tid][c]);
      const float corr = expf(mold - cm);
      float s = rsum[tid] * corr;
      for (int c = 0; c < 64; ++c) {
        const float p = expf(ks[tid][c] - cm);
        ks[tid][c] = p;
        s += p;
      }
      rmax[tid] = cm;
      rsum[tid] = s;
      rcorr[tid] = corr;
    }
    __syncthreads();
    // rescale accumulators, then acc += P @ V (contract over v)
#pragma unroll
    for (int i = 0; i < 8; ++i) {
      const float corr = rcorr[tr * 16 + i + half * 8];
      a0[i] *= corr;
      a1[i] *= corr;
    }
#pragma unroll
    for (int kk = 0; kk < 64; kk += 4) {
      const int ko = kk + 2 * half;
      const v2f a = *(const v2f*)&ks[tr * 16 + l16][ko];    // P row pair
      const v2f b0 = *(const v2f*)&vst[tc0 * 16 + l16][ko]; // V^T row pair
      const v2f b1 = *(const v2f*)&vst[tc0 * 16 + 16 + l16][ko];
      a0 = wmma_f32(a, b0, a0);
      a1 = wmma_f32(a, b1, a1);
    }
  }
  __syncthreads();
#pragma unroll
  for (int i = 0; i < 8; ++i) {
    const int row = tr * 16 + i + half * 8;  // slot
    const float den = rsum[row] + 1e-8f;
    const int c0i = tc0 * 16 + l16, c1i = c0i + 16;
    mout[((size_t)b * K_ + row) * D_ + h * DK_ + c0i] = a0[i] / den;
    mout[((size_t)b * K_ + row) * D_ + h * DK_ + c1i] = a1[i] / den;
  }
}

// ---------------------------------------------------------------------------
// empty-slot fallback: m[b,k,:] = cls[k,:] where empty
// ---------------------------------------------------------------------------
__global__ __launch_bounds__(256) void fixup_kernel(
    const float* __restrict__ emptyf, const float* __restrict__ cls,
    float* __restrict__ m) {
  const int idx = blockIdx.x * 256 + threadIdx.x;  // B*K*D
  const int d = idx & (D_ - 1);
  const int bk = idx >> 9;
  const int k = bk & (K_ - 1);
  if (emptyf[bk] > 0.5f) m[idx] = cls[k * D_ + d];
}

// ---------------------------------------------------------------------------
extern "C" void kernel_launch(void* const* d_in, const int* in_sizes, int n_in,
                              void* d_out, int out_size, void* d_ws,
                              size_t ws_size, hipStream_t stream) {
  const float* x      = (const float*)d_in[0];
  const float* assign = (const float*)d_in[1];
  const float* E      = (const float*)d_in[2];
  const float* cls0   = (const float*)d_in[3];
  const float* E2Dw   = (const float*)d_in[4];
  const float* E2Db   = (const float*)d_in[5];
  const float* Wq     = (const float*)d_in[6];
  const float* Wqb    = (const float*)d_in[7];
  const float* Wk     = (const float*)d_in[8];
  const float* Wkb    = (const float*)d_in[9];
  const float* Wv     = (const float*)d_in[10];
  const float* Wvb    = (const float*)d_in[11];
  const float* lng    = (const float*)d_in[12];
  const float* lnb    = (const float*)d_in[13];
  const float* fc1w   = (const float*)d_in[14];
  const float* fc1b   = (const float*)d_in[15];
  const float* fc2w   = (const float*)d_in[16];
  const float* fc2b   = (const float*)d_in[17];

  float* w = (float*)d_ws;
  float* cls    = w;  w += (size_t)K_ * D_;
  float* qn     = w;  w += (size_t)K_ * D_;
  float* q      = w;  w += (size_t)K_ * D_;
  float* kbuf   = w;  w += (size_t)B_ * V_ * D_;
  float* vbuf   = w;  w += (size_t)B_ * V_ * D_;
  float* mbuf   = w;  w += (size_t)B_ * K_ * D_;
  float* hbuf   = w;  w += (size_t)B_ * K_ * D_;
  float* ff1    = w;  w += (size_t)B_ * K_ * DFF_;
  float* emptyf = w;  w += (size_t)B_ * K_;

  float* out_m   = (float*)d_out;
  float* out_cnt = out_m + (size_t)B_ * K_ * D_;
  float* out_emp = out_cnt + (size_t)B_ * K_;

  // cls + LN(cls)
  cls_ln_kernel<<<K_, 256, 0, stream>>>(E, E2Dw, E2Db, cls0, lng, lnb, cls, qn);
  // q = LN(cls) @ Wq + b   [64,512]
  gemm_f32_wmma<0><<<dim3(D_ / 64, 1), 256, 0, stream>>>(
      qn, Wq, Wqb, nullptr, q, K_, D_, D_);
  // k, v projections  [65536,512] @ [512,512]
  gemm_f32_wmma<0><<<dim3(D_ / 64, (B_ * V_) / 128), 256, 0, stream>>>(
      x, Wk, Wkb, nullptr, kbuf, B_ * V_, D_, D_);
  gemm_f32_wmma<0><<<dim3(D_ / 64, (B_ * V_) / 128), 256, 0, stream>>>(
      x, Wv, Wvb, nullptr, vbuf, B_ * V_, D_, D_);
  // counts / empty mask
  count_kernel<<<B_, 256, 0, stream>>>(assign, out_cnt, out_emp, emptyf);
  // masked flash attention -> m
  attn_flash_kernel<<<B_ * H_, 256, 0, stream>>>(q, kbuf, vbuf, assign, mbuf);
  // empty fallback
  fixup_kernel<<<(B_ * K_ * D_) / 256, 256, 0, stream>>>(emptyf, cls, mbuf);
  // FFN: h = LN(m); out = m + gelu(h@fc1+b1)@fc2+b2
  ln_rows_kernel<<<B_ * K_, 256, 0, stream>>>(mbuf, lng, lnb, hbuf);
  gemm_f32_wmma<1><<<dim3(DFF_ / 64, (B_ * K_) / 128), 256, 0, stream>>>(
      hbuf, fc1w, fc1b, nullptr, ff1, B_ * K_, DFF_, D_);
  gemm_f32_wmma<2><<<dim3(D_ / 64, (B_ * K_) / 128), 256, 0, stream>>>(
      ff1, fc2w, fc2b, mbuf, out_m, B_ * K_, D_, DFF_);
}